// VQVAE_33818572488969
// MI455X (gfx1250) — compile-verified
//
#include <hip/hip_runtime.h>
#include <hip/hip_bf16.h>

typedef float v2f __attribute__((ext_vector_type(2)));
typedef float v8f __attribute__((ext_vector_type(8)));

constexpr int BATCH = 2048;
constexpr int DC    = 64;
constexpr int KDICT = 512;
constexpr int EMB   = 64;
constexpr float DECAYF    = 0.999f;
constexpr float ONE_MINUS = 0.001f;   // float(1.0 - 0.999) rounds to same value as 0.001f

// ---------------------------------------------------------------------------
// zero int scratch
__global__ void vq_zero_kernel(int* __restrict__ p, int n) {
    int i = blockIdx.x * blockDim.x + threadIdx.x;
    if (i < n) p[i] = 0;
}

// ---------------------------------------------------------------------------
// WMMA distance + argmin + histogram.
// grid.x = DC * 16  (c = blockIdx.x>>4, b-tile of 128 rows = blockIdx.x&15)
// block = 256 threads = 8 waves; wave w handles rows bbase + w*16 .. +15.
// LDS: dict[c] in pair-interleaved layout sd[(e>>1)*1024 + k*2 + (e&1)],
//      per-k squared norms, and a 512-bin int histogram.
__global__ __launch_bounds__(256)
void vq_argmin_kernel(const float* __restrict__ mu,
                      const float* __restrict__ dict,
                      int* __restrict__ idx_out,
                      int* __restrict__ hist) {
    extern __shared__ float smem[];
    float* sdict = smem;                    // KDICT*EMB floats
    float* sd2   = smem + KDICT * EMB;      // KDICT floats
    int*   shist = (int*)(sd2 + KDICT);     // KDICT ints

    const int c   = blockIdx.x >> 4;
    const int bt  = blockIdx.x & 15;
    const int tid = threadIdx.x;

    // ---- stage dictionary slice for this c into LDS, zero histogram ----
    const float* dc = dict + (size_t)c * KDICT * EMB;
    for (int i = tid; i < KDICT * EMB; i += 256) {
        int k = i >> 6;          // row
        int e = i & 63;          // element (coalesced global read)
        sdict[((e >> 1) << 10) + (k << 1) + (e & 1)] = dc[i];
    }
    for (int k = tid; k < KDICT; k += 256) shist[k] = 0;
    __syncthreads();

    // ---- per-k squared norms ----
    for (int k = tid; k < KDICT; k += 256) {
        float s = 0.0f;
        #pragma unroll 8
        for (int e = 0; e < EMB; ++e) {
            float v = sdict[((e >> 1) << 10) + (k << 1) + (e & 1)];
            s += v * v;
        }
        sd2[k] = s;
    }
    __syncthreads();

    // ---- WMMA score + argmin ----
    const int lane = tid & 31;
    const int wv   = tid >> 5;
    const int m    = lane & 15;            // A-row / C-column (N) index
    const int kk   = (lane >> 4) << 1;     // K-pair select: 0 or 2
    const int bbase = bt * 128 + wv * 16;

    // A tile: 16 rows x 64 cols of mu3[:,c,:], per ISA A-matrix 16x4 layout
    v2f a[16];
    const float* ap = mu + (size_t)(bbase + m) * (DC * EMB) + c * EMB;
    #pragma unroll
    for (int ch = 0; ch < 16; ++ch)
        a[ch] = *(const v2f*)(ap + ch * 4 + kk);

    float bestd[8];
    int   bestk[8];
    #pragma unroll
    for (int i = 0; i < 8; ++i) { bestd[i] = __builtin_inff(); bestk[i] = 0; }

    for (int kt = 0; kt < KDICT / 16; ++kt) {
        const int kcol = kt * 16 + m;      // global k handled by this lane slot
        v8f acc = {0.f, 0.f, 0.f, 0.f, 0.f, 0.f, 0.f, 0.f};
        #pragma unroll
        for (int ch = 0; ch < 16; ++ch) {
            const int e0 = ch * 4 + kk;    // even -> pair is adjacent in LDS
            v2f bmat = *(const v2f*)(&sdict[((e0 >> 1) << 10) + (kcol << 1)]);
            acc = __builtin_amdgcn_wmma_f32_16x16x4_f32(
                false, a[ch], false, bmat, (short)0, acc, false, false);
        }
        const float d2k = sd2[kcol];
        #pragma unroll
        for (int i = 0; i < 8; ++i) {
            float d = d2k - 2.0f * acc[i];         // ||d||^2 - 2 mu.d (argmin-equivalent)
            if (d < bestd[i]) { bestd[i] = d; bestk[i] = kcol; }
        }
    }

    // ---- reduce argmin across the 16 lanes of each half-wave ----
    #pragma unroll
    for (int i = 0; i < 8; ++i) {
        float d = bestd[i];
        int   k = bestk[i];
        #pragma unroll
        for (int mask = 8; mask >= 1; mask >>= 1) {
            float od = __shfl_xor(d, mask, 32);
            int   ok = __shfl_xor(k, mask, 32);
            if (od < d || (od == d && ok < k)) { d = od; k = ok; }  // first-min tie-break
        }
        if (m == 0) {
            int row = bbase + ((lane >> 4) << 3) + i;  // C layout: M = i (+8 for hi half)
            idx_out[row * DC + c] = k;
            atomicAdd(&shist[k], 1);
        }
    }
    __syncthreads();

    for (int k = tid; k < KDICT; k += 256) {
        int v = shist[k];
        if (v) atomicAdd(&hist[c * KDICT + k], v);
    }
}

// ---------------------------------------------------------------------------
__global__ void vq_counts_kernel(const float* __restrict__ ema,
                                 const int* __restrict__ hist,
                                 float* __restrict__ counts_new, int n) {
    int i = blockIdx.x * blockDim.x + threadIdx.x;
    if (i < n) counts_new[i] = DECAYF * ema[i] + ONE_MINUS * (float)hist[i];
}

// cw_embed gather: one thread per (b, c, e/4) -> float4
__global__ void vq_embed_kernel(const float* __restrict__ dict,
                                const int* __restrict__ idx,
                                float* __restrict__ cw, int n4) {
    int i = blockIdx.x * blockDim.x + threadIdx.x;
    if (i >= n4) return;
    int e4 = i & 15;
    int c  = (i >> 4) & 63;
    int b  = i >> 10;
    int k  = idx[b * DC + c];
    const float4* src = (const float4*)(dict + ((size_t)c * KDICT + k) * EMB) + e4;
    float4 v = *src;
    ((float4*)(cw + (size_t)b * DC * EMB + c * EMB))[e4] = v;
}

// one_hot: one thread per (b, c, k/4) -> float4 compare-store (pure streaming)
__global__ void vq_onehot_kernel(const int* __restrict__ idx,
                                 float* __restrict__ oh, int n4) {
    int i = blockIdx.x * blockDim.x + threadIdx.x;
    if (i >= n4) return;
    int k4 = i & 127;
    int c  = (i >> 7) & 63;
    int b  = i >> 13;
    int k  = idx[b * DC + c];
    int kb = k4 << 2;
    float4 v;
    v.x = (kb + 0 == k) ? 1.0f : 0.0f;
    v.y = (kb + 1 == k) ? 1.0f : 0.0f;
    v.z = (kb + 2 == k) ? 1.0f : 0.0f;
    v.w = (kb + 3 == k) ? 1.0f : 0.0f;
    ((float4*)oh)[i] = v;   // flat index matches (b*DC+c)*KDICT/4 + k4
}

__global__ void vq_dictinit_kernel(const float* __restrict__ dict,
                                   float* __restrict__ dnew, int n) {
    int i = blockIdx.x * blockDim.x + threadIdx.x;
    if (i < n) dnew[i] = DECAYF * dict[i];
}

// scatter-add contrib into dict_new via hardware fp32 global atomics
__global__ void vq_scatter_kernel(const float* __restrict__ mu,
                                  const int* __restrict__ idx,
                                  const float* __restrict__ counts_new,
                                  float* __restrict__ dnew, int n) {
    int i = blockIdx.x * blockDim.x + threadIdx.x;
    if (i >= n) return;                   // i = b*4096 + c*64 + e  (== mu flat index)
    int e = i & 63;
    int c = (i >> 6) & 63;
    int b = i >> 12;
    int k = idx[b * DC + c];
    float g = counts_new[c * KDICT + k];
    float v = ONE_MINUS * mu[i] / g;
    unsafeAtomicAdd(&dnew[((size_t)c * KDICT + k) * EMB + e], v);
}

// ---------------------------------------------------------------------------
extern "C" void kernel_launch(void* const* d_in, const int* in_sizes, int n_in,
                              void* d_out, int out_size, void* d_ws, size_t ws_size,
                              hipStream_t stream) {
    const float* mu   = (const float*)d_in[0];
    const float* dict = (const float*)d_in[1];
    const float* ema  = (const float*)d_in[2];

    float* out  = (float*)d_out;
    float* cw   = out;                                      // 2048*4096
    float* oh   = out + (size_t)BATCH * DC * EMB;           // +8388608: 2048*64*512
    float* dnew = oh  + (size_t)BATCH * DC * KDICT;         // +67108864: 64*512*64
    float* cnts = dnew + (size_t)DC * KDICT * EMB;          // +2097152: 64*512

    int* idx  = (int*)d_ws;                                 // BATCH*DC ints
    int* hist = idx + BATCH * DC;                           // DC*KDICT ints

    const int nhist = DC * KDICT;
    vq_zero_kernel<<<(nhist + 255) / 256, 256, 0, stream>>>(hist, nhist);

    size_t shmem = (size_t)(KDICT * EMB + KDICT) * sizeof(float) + KDICT * sizeof(int);
    vq_argmin_kernel<<<DC * 16, 256, shmem, stream>>>(mu, dict, idx, hist);

    vq_counts_kernel<<<(nhist + 255) / 256, 256, 0, stream>>>(ema, hist, cnts, nhist);

    int nEmb4 = BATCH * DC * (EMB / 4);
    vq_embed_kernel<<<(nEmb4 + 255) / 256, 256, 0, stream>>>(dict, idx, cw, nEmb4);

    int nOh4 = BATCH * DC * (KDICT / 4);
    vq_onehot_kernel<<<(nOh4 + 255) / 256, 256, 0, stream>>>(idx, oh, nOh4);

    int nDict = DC * KDICT * EMB;
    vq_dictinit_kernel<<<(nDict + 255) / 256, 256, 0, stream>>>(dict, dnew, nDict);

    int nContrib = BATCH * DC * EMB;
    vq_scatter_kernel<<<(nContrib + 255) / 256, 256, 0, stream>>>(mu, idx, cnts, dnew, nContrib);
}